// EdgePredictor_9010841387501
// MI455X (gfx1250) — compile-verified
//
#include <hip/hip_runtime.h>

// Problem constants (reference: B=2, S=1024, D=128)
#define SDIM 1024
#define DDIM 128
#define BATCH 2
#define MTOT (BATCH * SDIM)   // 2048 rows total

typedef float v8f __attribute__((ext_vector_type(8)));
typedef float v2f __attribute__((ext_vector_type(2)));

// ---------------------------------------------------------------------------
// Phase 1: projections via V_WMMA_F32_16X16X4_F32.
//   hi[m,d] = sum_k X[m,k] * W1[k,d]      + b1[d]   (bias folded in here)
//   hj[m,d] = sum_k X[m,k] * W1[128+k,d]
// One wave per 16x16 output tile; K accumulated in 32 WMMA steps of 4.
// A (16x4 f32): lanes 0-15 = M rows; VGPR0 holds K=(0|2), VGPR1 K=(1|3)
//   depending on lane>=16.  B (4x16): lanes = N cols, same K split.
// C/D (16x16 f32, 8 VGPRs): VGPR r -> M = r + (lane>=16 ? 8 : 0), N = lane&15.
// ---------------------------------------------------------------------------
__global__ __launch_bounds__(128) void proj_wmma_kernel(
    const float* __restrict__ X,    // (2048,128)
    const float* __restrict__ W1,   // (256,128) row-major (k, d)
    const float* __restrict__ b1,   // (128)
    float* __restrict__ hi,         // (2048,128)
    float* __restrict__ hj)         // (2048,128)
{
    const int lane    = threadIdx.x & 31;
    const int wave_id = blockIdx.x * 4 + (threadIdx.x >> 5);   // 0..2047
    const int which   = wave_id >> 10;     // 0 -> hi, 1 -> hj
    const int tile    = wave_id & 1023;
    const int mtile   = tile >> 3;         // 0..127  (M/16)
    const int ntile   = tile & 7;          // 0..7    (D/16)

    const int row  = mtile * 16 + (lane & 15);
    const int col  = ntile * 16 + (lane & 15);
    const int koff = (lane >> 4) * 2;      // K sub-offset per lane half

    const float* __restrict__ Wb  = W1 + which * DDIM * DDIM;  // half of W1
    float* __restrict__ out       = which ? hj : hi;

    v8f c = {};
    #pragma unroll 4
    for (int kk = 0; kk < DDIM; kk += 4) {
        // A: two consecutive K along the X row -> one b64 load
        v2f a = *(const v2f*)(X + row * DDIM + kk + koff);
        // B: two K rows of W1, same column
        v2f b;
        b.x = Wb[(kk + koff + 0) * DDIM + col];
        b.y = Wb[(kk + koff + 1) * DDIM + col];
        c = __builtin_amdgcn_wmma_f32_16x16x4_f32(
                /*neg_a=*/false, a, /*neg_b=*/false, b,
                /*c_mod=*/(short)0, c, /*reuse_a=*/false, /*reuse_b=*/false);
    }

    const float bias = which ? 0.0f : b1[col];
    const int mo = (lane >> 4) * 8;
    #pragma unroll
    for (int r = 0; r < 8; ++r) {
        out[(mtile * 16 + mo + r) * DDIM + col] = c[r] + bias;
    }
}

// ---------------------------------------------------------------------------
// Phase 2: out[b,i,j] = sum_d relu(hi[b*S+i,d] + hj[b*S+j,d]) * w2[d] + b2
// 64(i) x 32(j) tile per 256-thread block; each thread owns a 4x2 micro-tile
// (i: il, il+16, il+32, il+48 ; j: jl, jl+16).
// Both LDS tiles padded to 33-float4 row stride: all ds_read_b128 in the
// inner loop are bank-conflict-free (start bank = 4*row + 4*d4 mod 64).
// ---------------------------------------------------------------------------
__device__ __forceinline__ void rfma4(float4& acc, const float4 p,
                                      const float4 q, const float4 w) {
    acc.x = fmaf(fmaxf(p.x + q.x, 0.f), w.x, acc.x);
    acc.y = fmaf(fmaxf(p.y + q.y, 0.f), w.y, acc.y);
    acc.z = fmaf(fmaxf(p.z + q.z, 0.f), w.z, acc.z);
    acc.w = fmaf(fmaxf(p.w + q.w, 0.f), w.w, acc.w);
}

__global__ __launch_bounds__(256) void edge_pair_kernel(
    const float* __restrict__ hi,
    const float* __restrict__ hj,
    const float* __restrict__ W2,   // (128)
    const float* __restrict__ b2,   // (1)
    float* __restrict__ out)        // (B,S,S)
{
    __shared__ float4 his[64 * 33];   // 33,792 B
    __shared__ float4 hjs[32 * 33];   // 16,896 B
    __shared__ float4 w2s[32];        //    512 B   (total ~51 KB)

    const int t  = threadIdx.x;
    const int jt = blockIdx.x;      // 0..31  (j tile of 32)
    const int it = blockIdx.y;      // 0..15  (i tile of 64)
    const int bb = blockIdx.z;      // 0..1

    const float4* __restrict__ hig = (const float4*)(hi + (bb * SDIM + it * 64) * DDIM);
    const float4* __restrict__ hjg = (const float4*)(hj + (bb * SDIM + jt * 32) * DDIM);

    #pragma unroll
    for (int e = t; e < 64 * 32; e += 256) {      // 8 iters
        const int r = e >> 5, c4 = e & 31;
        his[r * 33 + c4] = hig[r * 32 + c4];
    }
    #pragma unroll
    for (int e = t; e < 32 * 32; e += 256) {      // 4 iters
        const int r = e >> 5, c4 = e & 31;
        hjs[r * 33 + c4] = hjg[r * 32 + c4];
    }
    if (t < 32) w2s[t] = ((const float4*)W2)[t];
    __syncthreads();

    const int jl = t & 15;          // j: jl, jl+16
    const int il = t >> 4;          // i: il, il+16, il+32, il+48

    float4 acc[4][2];
    #pragma unroll
    for (int a = 0; a < 4; ++a)
        #pragma unroll
        for (int b = 0; b < 2; ++b)
            acc[a][b] = make_float4(0.f, 0.f, 0.f, 0.f);

    #pragma unroll 4
    for (int d4 = 0; d4 < 32; ++d4) {
        const float4 w  = w2s[d4];
        const float4 p0 = his[(il +  0) * 33 + d4];
        const float4 p1 = his[(il + 16) * 33 + d4];
        const float4 p2 = his[(il + 32) * 33 + d4];
        const float4 p3 = his[(il + 48) * 33 + d4];
        const float4 q0 = hjs[(jl +  0) * 33 + d4];
        const float4 q1 = hjs[(jl + 16) * 33 + d4];

        rfma4(acc[0][0], p0, q0, w);  rfma4(acc[0][1], p0, q1, w);
        rfma4(acc[1][0], p1, q0, w);  rfma4(acc[1][1], p1, q1, w);
        rfma4(acc[2][0], p2, q0, w);  rfma4(acc[2][1], p2, q1, w);
        rfma4(acc[3][0], p3, q0, w);  rfma4(acc[3][1], p3, q1, w);
    }

    const float bias2 = b2[0];
    float* __restrict__ obase =
        out + (size_t)bb * SDIM * SDIM + (size_t)(it * 64) * SDIM + jt * 32;

    #pragma unroll
    for (int a = 0; a < 4; ++a) {
        #pragma unroll
        for (int b = 0; b < 2; ++b) {
            const float4 v = acc[a][b];
            const float r = bias2 + v.x + v.y + v.z + v.w;
            obase[(size_t)(il + 16 * a) * SDIM + jl + 16 * b] = r;
        }
    }
}

// ---------------------------------------------------------------------------
extern "C" void kernel_launch(void* const* d_in, const int* in_sizes, int n_in,
                              void* d_out, int out_size, void* d_ws, size_t ws_size,
                              hipStream_t stream) {
    const float* X  = (const float*)d_in[0];  // (2,1024,128)
    const float* W1 = (const float*)d_in[1];  // (256,128)
    const float* b1 = (const float*)d_in[2];  // (128)
    const float* W2 = (const float*)d_in[3];  // (128,1)
    const float* b2 = (const float*)d_in[4];  // (1)
    float* out = (float*)d_out;               // (2,1024,1024)

    float* hi = (float*)d_ws;                 // 2048*128 f32 = 1 MB
    float* hj = hi + MTOT * DDIM;             // next 1 MB (needs 2 MB of ws)

    // Phase 1: 2048 tiles (128 M-tiles x 8 N-tiles x 2 halves), 1 wave each.
    proj_wmma_kernel<<<512, 128, 0, stream>>>(X, W1, b1, hi, hj);

    // Phase 2: 64x32 output tile per block over (S/32, S/64, B).
    dim3 grid(SDIM / 32, SDIM / 64, BATCH);
    edge_pair_kernel<<<grid, 256, 0, stream>>>(hi, hj, W2, b2, out);
}